// CrossMerge3D_26319559590242
// MI455X (gfx1250) — compile-verified
//
#include <hip/hip_runtime.h>

// CrossMerge3D for MI455X (gfx1250): pure bandwidth problem (~327 MB moved,
// ~14 us floor at 23.3 TB/s). No contraction dim -> WMMA inapplicable.
// Strategy: every global access is a contiguous per-wave b128 stream; the
// two transposed scan families are staged through a conflict-free padded
// LDS volume (CDNA5 WGP LDS), reusing one 135 KB buffer for both.

namespace cm3d {
constexpr int Dd     = 32;
constexpr int N3     = Dd * Dd * Dd;     // 32768 voxels per (b,c) volume
constexpr int CH     = N3 / 4;           // 8192 float4 chunks per volume
constexpr int TPB    = 1024;             // 32 wave32 waves per workgroup
constexpr int ITER   = CH / TPB;         // 8 chunks per thread
constexpr int CROW   = 33;               // padded j-row stride (floats), odd vs 64 banks
constexpr int CPLANE = 1057;             // padded i-plane stride (floats), odd vs 64 banks
constexpr int LVOL   = Dd * CPLANE;      // 33824 floats = 135,296 B of LDS
constexpr int Bb = 2, Ss = 12, Cc = 96;
constexpr long SSTR  = (long)Cc * N3;    // stride between scans s and s+1 (floats)
}

__global__ __launch_bounds__(cm3d::TPB)
void cross_merge3d_kernel(const float* __restrict__ ys, float* __restrict__ out)
{
    using namespace cm3d;
    __shared__ float lds[LVOL];          // one staged transposed volume (reused)

    const int bc = blockIdx.x;           // bc = b*96 + c
    const int b  = bc / Cc;
    const int c  = bc - b * Cc;
    const float* vol = ys + ((long)b * Ss * Cc + c) * (long)N3;  // scan s at vol + s*SSTR

    const float4* s0  = (const float4*)(vol + 0  * SSTR);
    const float4* s1  = (const float4*)(vol + 1  * SSTR);
    const float4* s2  = (const float4*)(vol + 2  * SSTR);
    const float4* s3  = (const float4*)(vol + 3  * SSTR);
    const float4* s4  = (const float4*)(vol + 4  * SSTR);
    const float4* s5  = (const float4*)(vol + 5  * SSTR);
    const float4* s6  = (const float4*)(vol + 6  * SSTR);
    const float4* s7  = (const float4*)(vol + 7  * SSTR);
    const float4* s8  = (const float4*)(vol + 8  * SSTR);
    const float4* s9  = (const float4*)(vol + 9  * SSTR);
    const float4* s10 = (const float4*)(vol + 10 * SSTR);
    const float4* s11 = (const float4*)(vol + 11 * SSTR);
    float4* o4 = (float4*)(out + (long)bc * N3);

    const int tid = threadIdx.x;

    // Warm the i-scan volumes (consumed in the accumulate phase) while the
    // J-stage loads are in flight: gfx1250 global_prefetch_b8, one 128B line
    // per lane, 1024 lines per volume == whole 128 KB volume per block.
    __builtin_prefetch((const float*)s0 + tid * 32, 0, 0);
    __builtin_prefetch((const float*)s1 + tid * 32, 0, 0);
    __builtin_prefetch((const float*)s2 + tid * 32, 0, 0);
    __builtin_prefetch((const float*)s3 + tid * 32, 0, 0);

    // ---- stage sum_j = y4 + y5 + flip(y6 + y7), stored at OUTPUT coords ----
    // source flat n = j*1024 + k*32 + i  (i fastest) -> lds[i][j][k]
    #pragma unroll
    for (int it = 0; it < ITER; ++it) {
        const int ch = tid + it * TPB;
        const int rc = CH - 1 - ch;                    // flipped chunk, components reversed
        const float4 a  = s4[ch];
        const float4 bq = s5[ch];
        const float4 r0 = s6[rc];
        const float4 r1 = s7[rc];
        const int n = ch << 2;
        const int i = n & 31, k = (n >> 5) & 31, j = n >> 10;
        const int ad = i * CPLANE + j * CROW + k;
        lds[ad             ] = a.x + bq.x + r0.w + r1.w;
        lds[ad +     CPLANE] = a.y + bq.y + r0.z + r1.z;
        lds[ad + 2 * CPLANE] = a.z + bq.z + r0.y + r1.y;
        lds[ad + 3 * CPLANE] = a.w + bq.w + r0.x + r1.x;
    }
    __syncthreads();

    // ---- accumulate i-scan terms (coalesced global) + sum_j (LDS) ----
    float4 acc[ITER];
    #pragma unroll
    for (int it = 0; it < ITER; ++it) {
        const int ch = tid + it * TPB;
        const int rc = CH - 1 - ch;
        const float4 a  = s0[ch];
        const float4 bq = s1[ch];
        const float4 r0 = s2[rc];
        const float4 r1 = s3[rc];
        const int n = ch << 2;
        const int k = n & 31, j = (n >> 5) & 31, i = n >> 10;
        const int ad = i * CPLANE + j * CROW + k;
        acc[it].x = a.x + bq.x + r0.w + r1.w + lds[ad    ];
        acc[it].y = a.y + bq.y + r0.z + r1.z + lds[ad + 1];
        acc[it].z = a.z + bq.z + r0.y + r1.y + lds[ad + 2];
        acc[it].w = a.w + bq.w + r0.x + r1.x + lds[ad + 3];
    }
    __syncthreads();   // all sum_j reads complete before the buffer is reused

    // ---- stage sum_k = y8 + y9 + flip(y10 + y11) at OUTPUT coords ----
    // source flat n = k*1024 + i*32 + j  (j fastest) -> lds[i][j][k]
    #pragma unroll
    for (int it = 0; it < ITER; ++it) {
        const int ch = tid + it * TPB;
        const int rc = CH - 1 - ch;
        const float4 a  = s8[ch];
        const float4 bq = s9[ch];
        const float4 r0 = s10[rc];
        const float4 r1 = s11[rc];
        const int n = ch << 2;
        const int j = n & 31, i = (n >> 5) & 31, k = n >> 10;
        const int ad = i * CPLANE + j * CROW + k;
        lds[ad           ] = a.x + bq.x + r0.w + r1.w;
        lds[ad +     CROW] = a.y + bq.y + r0.z + r1.z;
        lds[ad + 2 * CROW] = a.z + bq.z + r0.y + r1.y;
        lds[ad + 3 * CROW] = a.w + bq.w + r0.x + r1.x;
    }
    __syncthreads();

    // ---- add sum_k, scale by 1/12, coalesced b128 store ----
    constexpr float SCALE = 1.0f / 12.0f;
    #pragma unroll
    for (int it = 0; it < ITER; ++it) {
        const int ch = tid + it * TPB;
        const int n = ch << 2;
        const int k = n & 31, j = (n >> 5) & 31, i = n >> 10;
        const int ad = i * CPLANE + j * CROW + k;
        float4 o;
        o.x = (acc[it].x + lds[ad    ]) * SCALE;
        o.y = (acc[it].y + lds[ad + 1]) * SCALE;
        o.z = (acc[it].z + lds[ad + 2]) * SCALE;
        o.w = (acc[it].w + lds[ad + 3]) * SCALE;
        o4[ch] = o;
    }
}

extern "C" void kernel_launch(void* const* d_in, const int* in_sizes, int n_in,
                              void* d_out, int out_size, void* d_ws, size_t ws_size,
                              hipStream_t stream) {
    (void)in_sizes; (void)n_in; (void)out_size; (void)d_ws; (void)ws_size;
    const float* ys  = (const float*)d_in[0];
    float*       out = (float*)d_out;
    // one workgroup per (b, c) pair: 2 * 96 = 192 blocks, 1024 threads each
    cross_merge3d_kernel<<<dim3(cm3d::Bb * cm3d::Cc), dim3(cm3d::TPB), 0, stream>>>(ys, out);
}